// GCN_conf_85718957294354
// MI455X (gfx1250) — compile-verified
//
#include <hip/hip_runtime.h>
#include <hip/hip_bf16.h>
#include <float.h>

// GCN forward: h = relu(A_sp @ (h @ W)) x2 ; out = l2норm(h); probs = softmax(out@wc+bc)
// N=50000 (16*3125), D=96 (16*6), C=40, E=800000.

typedef __attribute__((ext_vector_type(2))) float v2f;
typedef __attribute__((ext_vector_type(8))) float v8f;

#define FEAT 96
#define NCLS 40
#define EPB  8   // edges per spmm block

__global__ void zero_kernel(float* __restrict__ p, int n) {
    int i = blockIdx.x * blockDim.x + threadIdx.x;
    if (i < n) p[i] = 0.0f;
}

// C[M,96] = (RELU_A ? relu(A) : A)[M,96] @ B[96,96], one wave per 16-row tile.
// Uses V_WMMA_F32_16X16X4_F32: A frag = 2 VGPRs (lanes 0-15: K=k0,k0+1; lanes 16-31: K=k0+2,k0+3),
// B frag mirrored, C/D = 8 VGPRs (vgpr r: M=r for lanes 0-15, M=r+8 for lanes 16-31).
template <bool RELU_A>
__global__ __launch_bounds__(32) void gemm96_wmma(const float* __restrict__ A,
                                                  const float* __restrict__ B,
                                                  float* __restrict__ C) {
    const int lane = threadIdx.x & 31;
    const int half = lane >> 4;       // 0 or 1
    const int l16  = lane & 15;
    const int mrow = blockIdx.x * 16 + l16;
    const int ksel = half * 2;

    const float* __restrict__ arow = A + (size_t)mrow * FEAT;

    v8f acc[6] = {};

    #pragma unroll
    for (int k0 = 0; k0 < FEAT; k0 += 4) {
        // CDNA5 prefetch path (global_prefetch_b8) for the next A chunk
        __builtin_prefetch(arow + k0 + 4, 0, 0);

        float a0 = arow[k0 + ksel];
        float a1 = arow[k0 + ksel + 1];
        if (RELU_A) { a0 = fmaxf(a0, 0.0f); a1 = fmaxf(a1, 0.0f); }
        v2f a; a.x = a0; a.y = a1;

        const float* __restrict__ brow0 = B + (size_t)(k0 + ksel) * FEAT;
        const float* __restrict__ brow1 = brow0 + FEAT;

        #pragma unroll
        for (int t = 0; t < 6; ++t) {
            int n = t * 16 + l16;
            v2f b; b.x = brow0[n]; b.y = brow1[n];
            acc[t] = __builtin_amdgcn_wmma_f32_16x16x4_f32(
                false, a, false, b, (short)0, acc[t], false, false);
        }
    }

    const int rbase = blockIdx.x * 16 + half * 8;
    #pragma unroll
    for (int t = 0; t < 6; ++t) {
        #pragma unroll
        for (int r = 0; r < 8; ++r) {
            C[(size_t)(rbase + r) * FEAT + t * 16 + l16] = acc[t][r];
        }
    }
}

// Out[row[e], :] += vals[e] * T[col[e], :]  (segment_sum via L2 float atomics).
// blockDim = 96 (one thread per feature), EPB edges per block; row/col/vals are
// block-uniform -> scalar loads.
__global__ __launch_bounds__(96) void spmm_scatter(const int* __restrict__ row,
                                                   const int* __restrict__ col,
                                                   const float* __restrict__ vals,
                                                   const float* __restrict__ T,
                                                   float* __restrict__ Out, int E) {
    const int d = threadIdx.x;
    int e    = blockIdx.x * EPB;
    int eend = e + EPB;
    if (eend > E) eend = E;
    for (; e < eend; ++e) {
        const int   r = row[e];
        const int   c = col[e];
        const float v = vals[e];
        atomicAdd(&Out[(size_t)r * FEAT + d], v * T[(size_t)c * FEAT + d]);
    }
}

// Per node: relu -> L2 normalize -> out; logits = out@wc+bc -> softmax -> probs.
// 4 waves per block, one node per wave (wave32).
__global__ __launch_bounds__(128) void finalize_kernel(const float* __restrict__ M2,
                                                       const float* __restrict__ wc,
                                                       const float* __restrict__ bc,
                                                       float* __restrict__ out,
                                                       float* __restrict__ probs,
                                                       int N) {
    __shared__ float sh[4 * FEAT];
    const int w    = threadIdx.x >> 5;
    const int lane = threadIdx.x & 31;
    const int i    = blockIdx.x * 4 + w;
    if (i >= N) return;

    const size_t base = (size_t)i * FEAT;
    float h0 = fmaxf(M2[base + lane +  0], 0.0f);
    float h1 = fmaxf(M2[base + lane + 32], 0.0f);
    float h2 = fmaxf(M2[base + lane + 64], 0.0f);

    float sq = h0 * h0 + h1 * h1 + h2 * h2;
    #pragma unroll
    for (int m = 16; m >= 1; m >>= 1) sq += __shfl_xor(sq, m, 32);

    const float s = rsqrtf(fmaxf(sq, 1e-12f));
    h0 *= s; h1 *= s; h2 *= s;

    out[base + lane +  0] = h0;
    out[base + lane + 32] = h1;
    out[base + lane + 64] = h2;
    sh[w * FEAT + lane +  0] = h0;
    sh[w * FEAT + lane + 32] = h1;
    sh[w * FEAT + lane + 64] = h2;
    __syncthreads();

    // logits: lane handles class c1 = lane (0..31) and c2 = 32 + (lane&7) (valid for lane<8)
    const float* __restrict__ hv = &sh[w * FEAT];
    const int c2 = 32 + (lane & 7);
    float l1 = bc[lane];
    float l2a = 0.0f;
    #pragma unroll 4
    for (int d = 0; d < FEAT; ++d) {
        const float hd = hv[d];
        l1  += hd * wc[(size_t)d * NCLS + lane];
        l2a += hd * wc[(size_t)d * NCLS + c2];
    }
    const float l2 = bc[c2] + l2a;

    float mm = fmaxf(l1, (lane < 8) ? l2 : -FLT_MAX);
    #pragma unroll
    for (int m = 16; m >= 1; m >>= 1) mm = fmaxf(mm, __shfl_xor(mm, m, 32));

    const float e1 = expf(l1 - mm);
    const float e2 = (lane < 8) ? expf(l2 - mm) : 0.0f;
    float ssum = e1 + e2;
    #pragma unroll
    for (int m = 16; m >= 1; m >>= 1) ssum += __shfl_xor(ssum, m, 32);

    const float inv = 1.0f / ssum;
    probs[(size_t)i * NCLS + lane] = e1 * inv;
    if (lane < 8) probs[(size_t)i * NCLS + c2] = e2 * inv;
}

extern "C" void kernel_launch(void* const* d_in, const int* in_sizes, int n_in,
                              void* d_out, int out_size, void* d_ws, size_t ws_size,
                              hipStream_t stream) {
    const int*   row  = (const int*)d_in[0];
    const int*   col  = (const int*)d_in[1];
    const float* vals = (const float*)d_in[2];
    const float* x    = (const float*)d_in[3];
    const float* w1   = (const float*)d_in[4];
    const float* w2   = (const float*)d_in[5];
    const float* wc   = (const float*)d_in[6];
    const float* bc   = (const float*)d_in[7];

    const int E = in_sizes[0];
    const int N = in_sizes[3] / FEAT;       // 50000, divisible by 16

    float* t = (float*)d_ws;                // [N,96] GEMM output
    float* m = t + (size_t)N * FEAT;        // [N,96] SpMM accumulator
    float* out   = (float*)d_out;           // [N,96]
    float* probs = out + (size_t)N * FEAT;  // [N,40]

    const int nm      = N * FEAT;
    const int mtiles  = N / 16;
    const int eblocks = (E + EPB - 1) / EPB;

    // layer 1: t = x @ w1 ; m = A_sp @ t
    gemm96_wmma<false><<<mtiles, 32, 0, stream>>>(x, w1, t);
    zero_kernel<<<(nm + 255) / 256, 256, 0, stream>>>(m, nm);
    spmm_scatter<<<eblocks, 96, 0, stream>>>(row, col, vals, t, m, E);

    // layer 2: t = relu(m) @ w2 ; m = A_sp @ t   (stream order lets us reuse m)
    gemm96_wmma<true><<<mtiles, 32, 0, stream>>>(m, w2, t);
    zero_kernel<<<(nm + 255) / 256, 256, 0, stream>>>(m, nm);
    spmm_scatter<<<eblocks, 96, 0, stream>>>(row, col, vals, t, m, E);

    // relu + l2-normalize + classifier + softmax
    finalize_kernel<<<(N + 3) / 4, 128, 0, stream>>>(m, wc, bc, out, probs, N);
}